// KCMemory_73890617361028
// MI455X (gfx1250) — compile-verified
//
#include <hip/hip_runtime.h>

// Problem constants (fixed by the reference)
#define T_DIM  8192
#define DK     1024
#define RANK   64
#define DV     128
#define CHUNK  64
#define NCHUNK (T_DIM / CHUNK)   // 128
#define STATES (RANK * DV)       // 8192
#define LAMF   0.001f
#define ALPHAF 0.999f

typedef float v2f __attribute__((ext_vector_type(2)));
typedef float v8f __attribute__((ext_vector_type(8)));

// ---------------------------------------------------------------------------
// K1: phi = relu(K @ W^T + b) via V_WMMA_F32_16X16X4_F32 (exact fp32 path).
// One wave (32 lanes) per 16x16 output tile; K-loop steps of 4.
// A (16x4 f32): lanes 0-15 hold K=k+0/k+1 (vgpr0/1) for M=lane,
//               lanes 16-31 hold K=k+2/k+3 for M=lane-16  -> contiguous float2.
// B (4x16 f32): mirrored layout, B[k][n] = W[n][k]        -> contiguous float2.
// ---------------------------------------------------------------------------
__global__ __launch_bounds__(32) void kcm_phi_wmma(
    const float* __restrict__ K, const float* __restrict__ W,
    const float* __restrict__ b, float* __restrict__ phi)
{
    const int lane = threadIdx.x;
    const int m0   = blockIdx.x * 16;      // T tile
    const int n0   = blockIdx.y * 16;      // rank tile
    const int half = lane >> 4;            // 0 or 1
    const int mn   = lane & 15;            // M row for A, N col for B

    const float* arow = K + (size_t)(m0 + mn) * DK + 2 * half;
    const float* brow = W + (size_t)(n0 + mn) * DK + 2 * half;

    v8f c = {};
    for (int k = 0; k < DK; k += 4) {
        v2f a  = *(const v2f*)(arow + k);
        v2f bb = *(const v2f*)(brow + k);
        c = __builtin_amdgcn_wmma_f32_16x16x4_f32(
                /*neg_a=*/false, a, /*neg_b=*/false, bb,
                /*c_mod=*/(short)0, c, /*reuse_a=*/false, /*reuse_b=*/false);
    }

    const float bias = b[n0 + mn];
#pragma unroll
    for (int i = 0; i < 8; ++i) {
        const int m = i + 8 * half;        // D layout: vgpr i -> M=i / M=i+8
        float p = c[i] + bias;
        p = p > 0.0f ? p : 0.0f;
        phi[(size_t)(m0 + m) * RANK + n0 + mn] = p;
    }
}

// ---------------------------------------------------------------------------
// K2: per-chunk scan with zero init -> chunk totals.
// grid (NCHUNK, STATES/256); thread = one (r,v) state of one chunk.
// ---------------------------------------------------------------------------
__global__ __launch_bounds__(256) void kcm_chunk_total(
    const float* __restrict__ phi, const float* __restrict__ V,
    float* __restrict__ Tot)
{
    const int c   = blockIdx.x;
    const int idx = blockIdx.y * 256 + threadIdx.x;   // 0..8191
    const int r   = idx >> 7;
    const int v   = idx & (DV - 1);
    const int t0  = c * CHUNK;

    float s = 0.0f;
    for (int t = 0; t < CHUNK; ++t) {
        const float p  = phi[(size_t)(t0 + t) * RANK + r];   // broadcast in L0/L2
        const float vv = V[(size_t)(t0 + t) * DV + v];       // coalesced
        s = ALPHAF * s + LAMF * p * vv;
    }
    Tot[(size_t)c * STATES + idx] = s;
}

// ---------------------------------------------------------------------------
// K3: combine chunk totals sequentially -> per-chunk initial states.
// Init[c] = state entering chunk c;  Init[0] = M0.
// ---------------------------------------------------------------------------
__global__ __launch_bounds__(256) void kcm_chunk_init(
    const float* __restrict__ Tot, const float* __restrict__ M0,
    float* __restrict__ Init)
{
    const int idx = blockIdx.x * 256 + threadIdx.x;   // 0..8191
    float aL = 1.0f;
    for (int i = 0; i < CHUNK; ++i) aL *= ALPHAF;     // alpha^CHUNK
    float s = M0[idx];
    for (int c = 0; c < NCHUNK; ++c) {
        Init[(size_t)c * STATES + idx] = s;
        s = aL * s + Tot[(size_t)c * STATES + idx];
    }
}

// ---------------------------------------------------------------------------
// K4: final scan. One WG = (one chunk) x (32 d_v columns) x (all 64 ranks).
// tid = cc + 32*rg : column = lane, rank-group = wave.
//   -> every M_seq global_store_b32 writes 32 consecutive floats = one full
//      128B line (this is the dominant 268MB stream; coalescing it matters).
// Each thread owns 8 states (r = 8*rg + j) for its column.
// v_hat rank-reduction spans the 8 waves -> staged in sRed LDS, reduced once
// per 16-timestep sub-tile (2 barriers / 16 steps).
// ---------------------------------------------------------------------------
__global__ __launch_bounds__(256) void kcm_final(
    const float* __restrict__ phi, const float* __restrict__ V,
    const float* __restrict__ Init,
    float* __restrict__ Mseq, float* __restrict__ vhat)
{
    __shared__ float4 sPhi4[CHUNK * RANK / 4];  // [t][r]       16 KB
    __shared__ float4 sVal4[CHUNK * 32 / 4];    // [t][cc]       8 KB
    __shared__ float  sRed[16 * 8 * 32];        // [tt][rg][cc] 16 KB
    float* sPhi = (float*)sPhi4;
    float* sVal = (float*)sVal4;

    const int c   = blockIdx.x;           // chunk
    const int cb  = blockIdx.y * 32;      // column base
    const int tid = threadIdx.x;
    const int t0  = c * CHUNK;

    // phi chunk is contiguous in memory: copy 4096 floats as float4 (b128).
    {
        const float4* src = (const float4*)(phi + (size_t)t0 * RANK);
        for (int i = tid; i < CHUNK * RANK / 4; i += 256)
            sPhi4[i] = src[i];
    }
    // values chunk: 32-float row segments, 16B-aligned -> float4 loads.
    for (int i = tid; i < CHUNK * 8; i += 256) {
        const int t = i >> 3, q = i & 7;
        sVal4[i] = *(const float4*)(V + (size_t)(t0 + t) * DV + cb + 4 * q);
    }
    __syncthreads();

    const int cc = tid & 31;   // column within block (lane)
    const int rg = tid >> 5;   // rank group (wave id)

    float st[8];
#pragma unroll
    for (int j = 0; j < 8; ++j)
        st[j] = Init[(size_t)c * STATES + (size_t)(rg * 8 + j) * DV + cb + cc];

    for (int tb = 0; tb < CHUNK; tb += 16) {
        for (int tt = 0; tt < 16; ++tt) {
            const int t = tb + tt;
            const float vv = sVal[t * 32 + cc];
            float* outcol = Mseq + (size_t)(t0 + t) * STATES + cb + cc;
            float partial = 0.0f;
#pragma unroll
            for (int j = 0; j < 8; ++j) {
                const float ph = sPhi[t * RANK + rg * 8 + j];  // LDS broadcast
                const float s  = ALPHAF * st[j] + LAMF * ph * vv;
                st[j] = s;
                outcol[(size_t)(rg * 8 + j) * DV] = s;         // full-line store
                partial += ph * s;
            }
            sRed[tt * 256 + rg * 32 + cc] = partial;           // conflict-free
        }
        __syncthreads();
        // reduce 8 rank-groups per (tt, column); 512 outputs / 256 threads.
        for (int i = tid; i < 16 * 32; i += 256) {
            const int tt = i >> 5, c2 = i & 31;
            float sum = 0.0f;
#pragma unroll
            for (int g = 0; g < 8; ++g)
                sum += sRed[tt * 256 + g * 32 + c2];
            vhat[(size_t)(t0 + tb + tt) * DV + cb + c2] = sum;
        }
        __syncthreads();   // protect sRed before next sub-tile overwrites it
    }
}

extern "C" void kernel_launch(void* const* d_in, const int* in_sizes, int n_in,
                              void* d_out, int out_size, void* d_ws, size_t ws_size,
                              hipStream_t stream) {
    const float* keys   = (const float*)d_in[0];  // (T, d_k)
    const float* values = (const float*)d_in[1];  // (T, d_v)
    const float* W_phi  = (const float*)d_in[2];  // (rank, d_k)
    const float* b_phi  = (const float*)d_in[3];  // (rank,)
    const float* M0     = (const float*)d_in[4];  // (1, rank, d_v)

    float* Mseq = (float*)d_out;                          // (T, rank, d_v)
    float* vhat = Mseq + (size_t)T_DIM * RANK * DV;       // (T, d_v)

    float* ws   = (float*)d_ws;
    float* phi  = ws;                                     // T*RANK      = 2 MB
    float* Tot  = phi + (size_t)T_DIM * RANK;             // NCHUNK*8192 = 4 MB
    float* Init = Tot + (size_t)NCHUNK * STATES;          // NCHUNK*8192 = 4 MB

    kcm_phi_wmma   <<<dim3(T_DIM / 16, RANK / 16), 32,  0, stream>>>(keys, W_phi, b_phi, phi);
    kcm_chunk_total<<<dim3(NCHUNK, STATES / 256),  256, 0, stream>>>(phi, values, Tot);
    kcm_chunk_init <<<STATES / 256,                256, 0, stream>>>(Tot, M0, Init);
    kcm_final      <<<dim3(NCHUNK, DV / 32),       256, 0, stream>>>(phi, values, Init, Mseq, vhat);
}